// MoERouter_74904229642472
// MI455X (gfx1250) — compile-verified
//
#include <hip/hip_runtime.h>
#include <hip/hip_bf16.h>
#include <stdint.h>

#define T_TOKENS   32768
#define E_EXPERTS  256
#define TOPK       8
#define TILE_TOK   32
#define TOK_PER_BLK 64                       // two TDM tiles per block
#define NB         (T_TOKENS / TOK_PER_BLK)  // 512 blocks
#define ENTRIES_PER_BLK (TOK_PER_BLK * TOPK) // 512 routed entries per block
#define TK         (T_TOKENS * TOPK)

typedef unsigned int v4u __attribute__((ext_vector_type(4)));
typedef int          v8i __attribute__((ext_vector_type(8)));
typedef int          v4i __attribute__((ext_vector_type(4)));

// ---------------------------------------------------------------------------
// TDM: DMA one [TILE_TOK x 256] fp32 tile from global memory into LDS.
// D# group0: count=1 | lds_addr | global_addr(57b) | type=2.
// D# group1: data_size=4B, tensor_dim0=256, tensor_dim1=32768,
//            tile_dim0=256, tile_dim1=32, tensor_dim0_stride=256.
// Groups 2/3 zero (2-D tile; tile_dim2=0 => unused).
// ---------------------------------------------------------------------------
__device__ __forceinline__ void tdm_load_tile(const float* gsrc, void* ldsDst) {
  uint64_t ga = (uint64_t)(uintptr_t)gsrc;
  uint32_t lds_off = (uint32_t)(uintptr_t)ldsDst;   // generic ptr low32 = LDS offset
  v4u g0;
  g0[0] = 1u;                                            // count=1, no gather
  g0[1] = lds_off;                                       // lds_addr
  g0[2] = (uint32_t)ga;                                  // global_addr[31:0]
  g0[3] = (uint32_t)((ga >> 32) & 0x1FFFFFFull) | (2u << 30); // addr[56:32] | type=2
  v8i g1;
  g1[0] = 0x00020000;             // data_size = 2 (4 bytes)
  g1[1] = (int)(256u << 16);      // tensor_dim0 = 256  (low16 @ [31:16])
  g1[2] = (int)(0x8000u << 16);   // dim0 hi16=0 ; tensor_dim1 lo16 = 32768
  g1[3] = (int)(256u << 16);      // dim1 hi16=0 ; tile_dim0 = 256
  g1[4] = TILE_TOK;               // tile_dim1 = 32 ; tile_dim2 = 0
  g1[5] = 256;                    // tensor_dim0_stride = 256 elements
  g1[6] = 0;
  g1[7] = 0;
  v4i gz = {0, 0, 0, 0};
#if __clang_major__ >= 23
  v8i gz8 = {0, 0, 0, 0, 0, 0, 0, 0};
  __builtin_amdgcn_tensor_load_to_lds(g0, g1, gz, gz, gz8, 0);
#else
  __builtin_amdgcn_tensor_load_to_lds(g0, g1, gz, gz, 0);
#endif
}

// ---------------------------------------------------------------------------
// Wave32 argmax butterfly step via DPP16 (pure VALU, no LDS-pipe traffic).
// Payload: (sel value, expert idx, sigmoid value); tie -> lower expert idx.
// ---------------------------------------------------------------------------
template <int CTRL>
__device__ __forceinline__ void bfly_dpp(float& bv, int& be, float& bs) {
  float ov = __int_as_float(
      __builtin_amdgcn_update_dpp(0, __float_as_int(bv), CTRL, 0xF, 0xF, true));
  int oe = __builtin_amdgcn_update_dpp(0, be, CTRL, 0xF, 0xF, true);
  float os = __int_as_float(
      __builtin_amdgcn_update_dpp(0, __float_as_int(bs), CTRL, 0xF, 0xF, true));
  if (ov > bv || (ov == bv && oe < be)) { bv = ov; be = oe; bs = os; }
}

// Cross-row (xor16) step: after the 4 DPP steps each 16-lane row is fully
// reduced, so reading any lane of the opposite row (sel = 0) is correct.
__device__ __forceinline__ void bfly_x16(float& bv, int& be, float& bs) {
  float ov = __int_as_float(__builtin_amdgcn_permlanex16(
      __float_as_int(bv), __float_as_int(bv), 0, 0, false, false));
  int oe = __builtin_amdgcn_permlanex16(be, be, 0, 0, false, false);
  float os = __int_as_float(__builtin_amdgcn_permlanex16(
      __float_as_int(bs), __float_as_int(bs), 0, 0, false, false));
  if (ov > bv || (ov == bv && oe < be)) { bv = ov; be = oe; bs = os; }
}

__device__ __forceinline__ void wave_argmax(float& bv, int& be, float& bs) {
  bfly_dpp<0xB1>(bv, be, bs);   // quad_perm [1,0,3,2] : xor 1
  bfly_dpp<0x4E>(bv, be, bs);   // quad_perm [2,3,0,1] : xor 2
  bfly_dpp<0x141>(bv, be, bs);  // row_half_mirror     : merge quads in 8
  bfly_dpp<0x140>(bv, be, bs);  // row_mirror          : merge 8s in 16
  bfly_x16(bv, be, bs);         // cross-row           : merge 16s in 32
}

// ---------------------------------------------------------------------------
// Kernel A: sigmoid + biased top-8 per token (one wave32 per token),
//           TDM double-buffered tile loads, per-block ordered rank pass.
// ---------------------------------------------------------------------------
__device__ __forceinline__ void process_tile(
    const float* __restrict__ lbuf, int tile, int blk, int wave, int lane,
    const float* bb, float* __restrict__ scoresOut, int* __restrict__ idxOut,
    int* idxLDS) {
#pragma unroll
  for (int i = 0; i < TILE_TOK / 8; ++i) {   // 8 waves -> 4 tokens per wave
    const int tloc = wave * (TILE_TOK / 8) + i;
    const int tok  = blk * TOK_PER_BLK + tile * TILE_TOK + tloc;

    float sv[8], selv[8];
#pragma unroll
    for (int j = 0; j < 8; ++j) {
      float x = lbuf[tloc * E_EXPERTS + j * 32 + lane];  // bank-conflict-free
      float s = 1.0f / (1.0f + __expf(-x));
      sv[j] = s;
      selv[j] = s + bb[j];
    }

    float rsum = 0.0f;
    int myIdx = 0;
    float myS = 0.0f;
#pragma unroll
    for (int k = 0; k < TOPK; ++k) {
      // lane-local argmax over 8 slots (strict > keeps lowest expert index)
      float bv = -1e30f; int bj = 0;
#pragma unroll
      for (int j = 0; j < 8; ++j)
        if (selv[j] > bv) { bv = selv[j]; bj = j; }
      int be = bj * 32 + lane;
      float bs = sv[0];
#pragma unroll
      for (int j = 1; j < 8; ++j) if (bj == j) bs = sv[j];

      wave_argmax(bv, be, bs);  // all lanes agree on winner afterwards

      rsum += bs;
      if (lane == k) { myIdx = be; myS = bs; }
      // knock out the winner
      int wj = be >> 5;
      if ((be & 31) == lane) {
#pragma unroll
        for (int j = 0; j < 8; ++j) if (j == wj) selv[j] = -1e30f;
      }
    }

    if (lane < TOPK) {
      scoresOut[tok * TOPK + lane] = myS / rsum;
      idxOut[tok * TOPK + lane] = myIdx;
      idxLDS[(tile * TILE_TOK + tloc) * TOPK + lane] = myIdx;
    }
  }
}

__global__ __launch_bounds__(256) void moe_topk_kernel(
    const float* __restrict__ logits, const float* __restrict__ bias,
    float* __restrict__ scoresOut, int* __restrict__ idxOut,
    int* __restrict__ countsWS, int* __restrict__ rankWS) {
  __shared__ float ldsBuf[2][TILE_TOK * E_EXPERTS];  // 2 x 32 KB
  __shared__ int   idxLDS[ENTRIES_PER_BLK];          // 2 KB

  const int tid  = threadIdx.x;
  const int lane = tid & 31;
  const int wave = tid >> 5;
  const int blk  = blockIdx.x;

  // per-lane correction bias for experts e = j*32 + lane
  float bb[8];
#pragma unroll
  for (int j = 0; j < 8; ++j) bb[j] = bias[j * 32 + lane];

  const float* gbase = logits + (size_t)blk * TOK_PER_BLK * E_EXPERTS;

  if (wave == 0) {
    tdm_load_tile(gbase, &ldsBuf[0][0]);                            // tile 0
    tdm_load_tile(gbase + TILE_TOK * E_EXPERTS, &ldsBuf[1][0]);     // tile 1
  }

  if (wave == 0) __builtin_amdgcn_s_wait_tensorcnt(1);  // tile 0 landed
  __syncthreads();
  process_tile(&ldsBuf[0][0], 0, blk, wave, lane, bb, scoresOut, idxOut, idxLDS);

  if (wave == 0) __builtin_amdgcn_s_wait_tensorcnt(0);  // tile 1 landed
  __syncthreads();
  process_tile(&ldsBuf[1][0], 1, blk, wave, lane, bb, scoresOut, idxOut, idxLDS);

  __syncthreads();  // idxLDS complete

  // Ordered rank pass: thread e (=tid, blockDim==E) scans the block's 512
  // entries in flat row-major order, 4 at a time (ds_load_b128 broadcasts).
  const int e = tid;
  const int4* idx4 = (const int4*)idxLDS;
  int c = 0;
  for (int p4 = 0; p4 < ENTRIES_PER_BLK / 4; ++p4) {
    int4 v = idx4[p4];
    const int pb = blk * ENTRIES_PER_BLK + p4 * 4;
    if (v.x == e) { rankWS[pb + 0] = c; ++c; }
    if (v.y == e) { rankWS[pb + 1] = c; ++c; }
    if (v.z == e) { rankWS[pb + 2] = c; ++c; }
    if (v.w == e) { rankWS[pb + 3] = c; ++c; }
  }
  countsWS[e * NB + blk] = c;
}

// ---------------------------------------------------------------------------
// Kernel B: one wave per expert; exclusive scan of 512 per-block counts
// (in place), total -> expert_counts. All 16 loads hoisted before stores.
// ---------------------------------------------------------------------------
__global__ __launch_bounds__(256) void moe_scan_kernel(
    int* __restrict__ countsWS, int* __restrict__ cntOut) {
  const int lane = threadIdx.x & 31;
  const int e = blockIdx.x * 8 + (threadIdx.x >> 5);

  int vals[NB / 32];
#pragma unroll
  for (int c = 0; c < NB / 32; ++c)
    vals[c] = countsWS[e * NB + c * 32 + lane];

  int base = 0;
#pragma unroll
  for (int c = 0; c < NB / 32; ++c) {
    int v = vals[c];
    int inc = v;
#pragma unroll
    for (int off = 1; off < 32; off <<= 1) {
      int o = __shfl_up(inc, off);
      if (lane >= off) inc += o;
    }
    countsWS[e * NB + c * 32 + lane] = base + (inc - v);  // exclusive base
    base += __shfl(inc, 31);
  }
  if (lane == 0) cntOut[e] = base;
}

// ---------------------------------------------------------------------------
// Kernel C: offsets = block base for (expert, block) + local rank.
// ---------------------------------------------------------------------------
__global__ __launch_bounds__(256) void moe_offsets_kernel(
    const int* __restrict__ idxOut, const int* __restrict__ baseWS,
    const int* __restrict__ rankWS, int* __restrict__ offOut) {
  const int t = blockIdx.x * 256 + threadIdx.x;  // 0..TK-1, == blk*512 + p
  const int b = t >> 9;                          // / ENTRIES_PER_BLK
  const int e = idxOut[t];
  offOut[t] = baseWS[e * NB + b] + rankWS[t];
}

// ---------------------------------------------------------------------------
// d_in: [0]=hidden_states (UNUSED), [1]=router_logits f32, [2]=bias f32,
//       [3]=top_k (scalar, ==8). d_out: scores|idx|expert_counts|offsets,
// all 4-byte elements, concatenated flat.
// ---------------------------------------------------------------------------
extern "C" void kernel_launch(void* const* d_in, const int* in_sizes, int n_in,
                              void* d_out, int out_size, void* d_ws, size_t ws_size,
                              hipStream_t stream) {
  const float* logits = (const float*)d_in[1];
  const float* bias   = (const float*)d_in[2];

  float* scoresOut = (float*)d_out;                    // TK f32
  int*   idxOut    = (int*)d_out + TK;                 // TK i32
  int*   cntOut    = (int*)d_out + 2 * TK;             // 256 i32
  int*   offOut    = (int*)d_out + 2 * TK + E_EXPERTS; // TK i32

  int* countsWS = (int*)d_ws;                  // E * NB = 512 KB
  int* rankWS   = countsWS + E_EXPERTS * NB;   // TK     = 1 MB

  moe_topk_kernel<<<NB, 256, 0, stream>>>(logits, bias, scoresOut, idxOut,
                                          countsWS, rankWS);
  moe_scan_kernel<<<E_EXPERTS / 8, 256, 0, stream>>>(countsWS, cntOut);
  moe_offsets_kernel<<<TK / 256, 256, 0, stream>>>(idxOut, countsWS, rankWS,
                                                   offOut);
}